// MaskFormerHungarianMatcher_1812476199684
// MI455X (gfx1250) — compile-verified
//
#include <hip/hip_runtime.h>
#include <hip/hip_bf16.h>
#include <math.h>

// Problem constants (from reference setup_inputs)
#define BB 2
#define QQ 100
#define CC 134
#define HH 256
#define WW 256
#define TT 20
#define HWK 65536
#define HM 512
#define WM 512
#define KSPLIT 16
#define KPER (HWK / KSPLIT)   // 4096
#define QTILES 7              // ceil(100/16)

typedef __attribute__((ext_vector_type(2))) float v2f;
typedef __attribute__((ext_vector_type(4))) float v4f;
typedef __attribute__((ext_vector_type(8))) float v8f;

static __device__ inline v8f wmma_f32_16x16x4(v2f a, v2f b, v8f c) {
  // D = A(16x4 f32) x B(4x16 f32) + C(16x16 f32)
  return __builtin_amdgcn_wmma_f32_16x16x4_f32(false, a, false, b, (short)0, c,
                                               false, false);
}

// ---------------------------------------------------------------------------
// Kernel 1: class cost. One block per (b,q): softmax over C, gather labels.
// cost[b][q][t] = -softmax(class_logits[b][q])[class_labels[b][t]]
// ---------------------------------------------------------------------------
__global__ __launch_bounds__(128) void class_cost_kernel(
    const float* __restrict__ logits, const int* __restrict__ labels,
    float* __restrict__ cost) {
  const int bq = blockIdx.x;  // 0..BB*QQ-1
  const int b = bq / QQ;
  const int tid = threadIdx.x;
  const float* lg = logits + (size_t)bq * CC;
  __shared__ float red[128];

  float m = -INFINITY;
  for (int c = tid; c < CC; c += 128) m = fmaxf(m, lg[c]);
  red[tid] = m;
  __syncthreads();
  for (int s = 64; s > 0; s >>= 1) {
    if (tid < s) red[tid] = fmaxf(red[tid], red[tid + s]);
    __syncthreads();
  }
  const float mx = red[0];
  __syncthreads();
  float sum = 0.f;
  for (int c = tid; c < CC; c += 128) sum += __expf(lg[c] - mx);
  red[tid] = sum;
  __syncthreads();
  for (int s = 64; s > 0; s >>= 1) {
    if (tid < s) red[tid] += red[tid + s];
    __syncthreads();
  }
  const float inv = __builtin_amdgcn_rcpf(red[0]);
  if (tid < TT) {
    const int lab = labels[b * TT + tid];
    const float prob = __expf(lg[lab] - mx) * inv;
    cost[(size_t)bq * TT + tid] = -prob;
  }
}

// ---------------------------------------------------------------------------
// Kernel 2: WMMA partial contraction kernel.
// grid = (QTILES, BB, KSPLIT), block = 256 (8 wave32).
// Branch-free inner loop; fast transcendentals (1 exp, 1 rcp, 1 log / elem).
// ---------------------------------------------------------------------------
__global__ __launch_bounds__(256) void partial_kernel(
    const float* __restrict__ masks, const float* __restrict__ mlabels,
    float* __restrict__ pG, float* __restrict__ pP, float* __restrict__ pF,
    float* __restrict__ pS, float* __restrict__ pL) {
  const int qtile = blockIdx.x;  // 0..6
  const int b = blockIdx.y;      // 0..1
  const int ks = blockIdx.z;     // 0..15
  const int tid = threadIdx.x;
  const int wave = tid >> 5;
  const int lane = tid & 31;
  const int row = lane & 15;   // M row within tile (A) / N col (B,C)
  const int half = lane >> 4;  // K-half selector for A and B
  const int q = qtile * 16 + row;
  const bool qok = (q < QQ);
  const float qm = qok ? 1.f : 0.f;
  const float* mrow = masks + (size_t)(b * QQ + (qok ? q : 0)) * HWK;
  const int t0 = row;       // tile0 target row
  const int t1 = 16 + row;  // tile1 target row
  const bool t1ok = (t1 < TT);
  const float t1m = t1ok ? 1.f : 0.f;
  const float* lrow0 = mlabels + (size_t)(b * TT + t0) * (HM * WM);
  const float* lrow1 = mlabels + (size_t)(b * TT + (t1ok ? t1 : 0)) * (HM * WM);

  v8f accG0 = {}, accG1 = {}, accP0 = {}, accP1 = {};
  float sF = 0.f, sP = 0.f, sL0 = 0.f, sL1 = 0.f;

  const int kbase = ks * KPER + wave * 4;
#pragma unroll 2
  for (int n = 0; n < KPER / 32; ++n) {
    const int kk = kbase + n * 32 + half * 2;  // even -> 8B/16B alignment

    // ---- A-side: one b64 load of {x0,x1}; focal transform; mask by qm ----
    const v2f x01 = *(const v2f*)(mrow + kk);
    float g[2], pv[2];
#pragma unroll
    for (int e = 0; e < 2; ++e) {
      const float x = x01[e];
      const float ax = fabsf(x);
      const float t = __expf(-ax);                    // exp(-|x|), in (0,1]
      const float r = __builtin_amdgcn_rcpf(1.f + t); // 1/(1+t)
      const float p = (x >= 0.f) ? r : t * r;         // sigmoid(x)
      const float l1p = __logf(1.f + t);              // log1p(exp(-|x|))
      const float cep = l1p + fmaxf(-x, 0.f);         // softplus(-x)
      const float cen = l1p + fmaxf(x, 0.f);          // softplus(x)
      const float omp = 1.f - p;
      const float fpos = 0.25f * omp * omp * cep;
      const float fneg = 0.75f * p * p * cen;
      g[e] = qm * (fpos - fneg);
      pv[e] = qm * p;
      sF += qm * fneg;
      sP += qm * p;
    }
    const v2f aG = {g[0], g[1]};
    const v2f aP = {pv[0], pv[1]};

    // ---- B-side: nearest-downsampled labels. kk even => w even, same row;
    //      elements at o0 and o0+2, 16B aligned -> one b128 load per t-row.
    const int h0 = kk >> 8, w0 = kk & 255;
    const size_t o0 = ((size_t)(h0 << 1)) * WM + (size_t)(w0 << 1);
    const v4f L0 = *(const v4f*)(lrow0 + o0);
    const v4f L1 = *(const v4f*)(lrow1 + o0);
    const float l00 = L0.x, l01 = L0.z;
    const float l10 = L1.x * t1m, l11 = L1.z * t1m;
    sL0 += l00 + l01;
    sL1 += l10 + l11;
    const v2f b0 = {l00, l01};
    const v2f b1 = {l10, l11};

    accG0 = wmma_f32_16x16x4(aG, b0, accG0);
    accG1 = wmma_f32_16x16x4(aG, b1, accG1);
    accP0 = wmma_f32_16x16x4(aP, b0, accP0);
    accP1 = wmma_f32_16x16x4(aP, b1, accP1);
  }

  // ---- deterministic cross-wave reduction via per-wave LDS slabs ----
  __shared__ float wG[8][2][16][16];
  __shared__ float wPm[8][2][16][16];
  __shared__ float wF[8][2][16];
  __shared__ float wS[8][2][16];
  __shared__ float wL[8][2][32];
  __shared__ float fnegR[16], psumR[16], labT[32];

  // C/D layout: lane<16 -> (M=r, N=lane); lane>=16 -> (M=8+r, N=lane-16)
  const int ncol = lane & 15;
  const int moff = half * 8;
#pragma unroll
  for (int r = 0; r < 8; ++r) {
    wG[wave][0][moff + r][ncol] = accG0[r];
    wG[wave][1][moff + r][ncol] = accG1[r];
    wPm[wave][0][moff + r][ncol] = accP0[r];
    wPm[wave][1][moff + r][ncol] = accP1[r];
  }
  wF[wave][half][row] = sF;
  wS[wave][half][row] = sP;
  wL[wave][half][t0] = sL0;
  wL[wave][half][t1] = sL1;
  __syncthreads();

  if (tid < 16) {
    float a = 0.f, s = 0.f;
    for (int w = 0; w < 8; ++w) {
      a += wF[w][0][tid] + wF[w][1][tid];
      s += wS[w][0][tid] + wS[w][1][tid];
    }
    fnegR[tid] = a;
    psumR[tid] = s;
  } else if (tid < 48) {
    const int t = tid - 16;
    float s = 0.f;
    for (int w = 0; w < 8; ++w) s += wL[w][0][t] + wL[w][1][t];
    labT[t] = s;
  }
  __syncthreads();

  for (int idx = tid; idx < 16 * TT; idx += 256) {
    const int r = idx / TT, t = idx % TT;
    const int qq = qtile * 16 + r;
    if (qq < QQ) {
      const int tile = t >> 4, tc = t & 15;
      float gsum = 0.f, psum = 0.f;
      for (int w = 0; w < 8; ++w) {
        gsum += wG[w][tile][r][tc];
        psum += wPm[w][tile][r][tc];
      }
      const size_t o = ((size_t)((ks * BB + b) * QQ + qq)) * TT + t;
      pG[o] = gsum;
      pP[o] = psum;
    }
  }
  if (tid < 16) {
    const int qq = qtile * 16 + tid;
    if (qq < QQ) {
      pF[(ks * BB + b) * QQ + qq] = fnegR[tid];
      pS[(ks * BB + b) * QQ + qq] = psumR[tid];
    }
  }
  if (tid >= 16 && tid < 16 + TT) {
    const int t = tid - 16;
    pL[(ks * BB + b) * TT + t] = labT[t];  // identical across qtiles: benign
  }
}

// ---------------------------------------------------------------------------
// Kernel 3: finalize cost matrix (sums the 16 K-splits in fixed order).
// ---------------------------------------------------------------------------
__global__ void finalize_kernel(float* __restrict__ cost,
                                const float* __restrict__ pG,
                                const float* __restrict__ pP,
                                const float* __restrict__ pF,
                                const float* __restrict__ pS,
                                const float* __restrict__ pL) {
  const int idx = blockIdx.x * blockDim.x + threadIdx.x;
  if (idx >= BB * QQ * TT) return;
  const int t = idx % TT;
  const int q = (idx / TT) % QQ;
  const int b = idx / (QQ * TT);
  float g = 0.f, pp = 0.f, fn = 0.f, ps = 0.f, ls = 0.f;
  for (int ks = 0; ks < KSPLIT; ++ks) {
    const size_t o = ((size_t)((ks * BB + b) * QQ + q)) * TT + t;
    g += pG[o];
    pp += pP[o];
    fn += pF[(ks * BB + b) * QQ + q];
    ps += pS[(ks * BB + b) * QQ + q];
    ls += pL[(ks * BB + b) * TT + t];
  }
  const float cmask = (g + fn) * (1.f / (float)HWK);
  const float denom = ps + ls;
  const float cdice = 1.f - (2.f * pp + 1.f) / (denom + 1.f);
  cost[idx] += cmask + cdice;  // cost already holds class cost
}

// ---------------------------------------------------------------------------
// Kernel 4: Hungarian (Jonker-Volgenant potentials), one block per batch.
// Square n=100; columns 21..100 are zero-cost dummies. Inner scan is
// parallelized across 128 threads with an LDS argmin reduction.
// ---------------------------------------------------------------------------
#define HN 100
__global__ __launch_bounds__(128) void hungarian_kernel(
    const float* __restrict__ cost, int* __restrict__ out) {
  const int b = blockIdx.x;
  const int tid = threadIdx.x;
  __shared__ float u[HN + 1], v[HN + 1], minv[HN + 1];
  __shared__ int p[HN + 1], way[HN + 1];
  __shared__ unsigned char used[HN + 1];
  __shared__ float sh_delta;
  __shared__ int sh_j0;
  __shared__ float rmin[128];
  __shared__ int rarg[128];

  for (int j = tid; j <= HN; j += 128) {
    u[j] = 0.f;
    v[j] = 0.f;
    p[j] = 0;
    way[j] = 0;
  }
  __syncthreads();

  for (int i = 1; i <= HN; ++i) {
    if (tid == 0) {
      p[0] = i;
      sh_j0 = 0;
    }
    for (int j = tid; j <= HN; j += 128) {
      minv[j] = INFINITY;
      used[j] = 0;
    }
    __syncthreads();

    while (true) {
      const int j0 = sh_j0;
      if (tid == 0) used[j0] = 1;
      __syncthreads();
      const int i0 = p[j0];
      const float ui0 = u[i0];

      float lmin = INFINITY;
      int larg = HN + 1;
      const int j = tid + 1;  // 128 threads cover j = 1..100
      if (j <= HN && !used[j]) {
        const float aij =
            (j <= TT) ? cost[(size_t)(b * QQ + (i0 - 1)) * TT + (j - 1)] : 0.f;
        const float cur = aij - ui0 - v[j];
        if (cur < minv[j]) {
          minv[j] = cur;
          way[j] = j0;
        }
        lmin = minv[j];
        larg = j;
      }
      rmin[tid] = lmin;
      rarg[tid] = larg;
      __syncthreads();
      for (int s = 64; s > 0; s >>= 1) {
        if (tid < s) {
          const bool take = (rmin[tid + s] < rmin[tid]) ||
                            (rmin[tid + s] == rmin[tid] &&
                             rarg[tid + s] < rarg[tid]);
          if (take) {
            rmin[tid] = rmin[tid + s];
            rarg[tid] = rarg[tid + s];
          }
        }
        __syncthreads();
      }
      if (tid == 0) {
        sh_delta = rmin[0];
        sh_j0 = rarg[0];
      }
      __syncthreads();
      const float delta = sh_delta;
      for (int jj = tid; jj <= HN; jj += 128) {
        if (used[jj]) {
          u[p[jj]] += delta;  // distinct rows among used cols: race-free
          v[jj] -= delta;
        } else {
          minv[jj] -= delta;
        }
      }
      __syncthreads();
      if (p[sh_j0] == 0) break;  // uniform across block
    }

    __syncthreads();
    if (tid == 0) {
      int j0 = sh_j0;
      while (j0) {
        const int j1 = way[j0];
        p[j0] = p[j1];
        j0 = j1;
      }
    }
    __syncthreads();
  }

  if (tid == 0) {
    int rows[TT], cols[TT];
    for (int j = 1; j <= TT; ++j) {
      rows[j - 1] = p[j] - 1;
      cols[j - 1] = j - 1;
    }
    // sort pairs by row ascending (matches reference emission order)
    for (int a = 1; a < TT; ++a) {
      const int r = rows[a], c = cols[a];
      int k = a - 1;
      while (k >= 0 && rows[k] > r) {
        rows[k + 1] = rows[k];
        cols[k + 1] = cols[k];
        --k;
      }
      rows[k + 1] = r;
      cols[k + 1] = c;
    }
    for (int k = 0; k < TT; ++k) {
      out[b * 2 * TT + k] = rows[k];
      out[b * 2 * TT + TT + k] = cols[k];
    }
  }
}

// ---------------------------------------------------------------------------
extern "C" void kernel_launch(void* const* d_in, const int* in_sizes, int n_in,
                              void* d_out, int out_size, void* d_ws,
                              size_t ws_size, hipStream_t stream) {
  const float* masks = (const float*)d_in[0];    // [B,Q,H,W] f32
  const float* clslog = (const float*)d_in[1];   // [B,Q,C]  f32
  const float* mlab = (const float*)d_in[2];     // [B,T,Hm,Wm] f32
  const int* clabels = (const int*)d_in[3];      // [B,T] i32

  float* ws = (float*)d_ws;
  float* cost = ws;                                // BB*QQ*TT            = 4000
  float* pG = cost + (BB * QQ * TT);               // KSPLIT*BB*QQ*TT     = 64000
  float* pP = pG + (KSPLIT * BB * QQ * TT);        // 64000
  float* pF = pP + (KSPLIT * BB * QQ * TT);        // KSPLIT*BB*QQ        = 3200
  float* pS = pF + (KSPLIT * BB * QQ);             // 3200
  float* pL = pS + (KSPLIT * BB * QQ);             // KSPLIT*BB*TT        = 640
  int* out = (int*)d_out;                          // [B,2,T] i32 = 80

  class_cost_kernel<<<BB * QQ, 128, 0, stream>>>(clslog, clabels, cost);

  dim3 grid2(QTILES, BB, KSPLIT);
  partial_kernel<<<grid2, 256, 0, stream>>>(masks, mlab, pG, pP, pF, pS, pL);

  finalize_kernel<<<(BB * QQ * TT + 255) / 256, 256, 0, stream>>>(cost, pG, pP,
                                                                  pF, pS, pL);

  hungarian_kernel<<<BB, 128, 0, stream>>>(cost, out);
}